// GetScalars_74363063763118
// MI455X (gfx1250) — compile-verified
//
#include <hip/hip_runtime.h>

// CDNA5 / gfx1250 — wave32, WMMA f32 16x16x4.
typedef __attribute__((ext_vector_type(2))) float v2f;
typedef __attribute__((ext_vector_type(8))) float v8f;

#define NB 4
#define NA 4096
#define NC 32
#define NATOMS (NB * NA)                  // 16384
#define OUT_STRIDE (NC + 4 * NC * NC)     // 32 + 4096 = 4128

// One wave32 per atom. 8 waves (256 threads) per block -> 8 atoms/block.
// Grid = 16384/8 = 2048 blocks exactly, so EXEC is all-1s in every wave
// (required for V_WMMA). No EXEC-divergent control flow anywhere below:
// K-padding is done with value-selects (v_cndmask) on unconditionally
// loaded, address-clamped data — never with guarded loads.
__global__ __launch_bounds__(256) void so3_gram_wmma_kernel(
    const float* __restrict__ p0,   // [atom][ 1*32]
    const float* __restrict__ p1,   // [atom][ 3*32]
    const float* __restrict__ p2,   // [atom][ 5*32]
    const float* __restrict__ p3,   // [atom][ 7*32]
    float* __restrict__ out)        // [atom][4128]
{
    const int lane = threadIdx.x & 31;
    const int wave = threadIdx.x >> 5;
    const int atom = blockIdx.x * 8 + wave;   // flat b*NA + a, always < NATOMS

    const int half = lane >> 4;   // which 16-lane half of the wave
    const int lm   = lane & 15;

    const float* __restrict__ parts[4] = {
        p0 + (size_t)atom * (1 * NC),
        p1 + (size_t)atom * (3 * NC),
        p2 + (size_t)atom * (5 * NC),
        p3 + (size_t)atom * (7 * NC),
    };
    float* __restrict__ o = out + (size_t)atom * OUT_STRIDE;

    // Prefetch the next atom's largest input block (clamped index: tail waves
    // redundantly prefetch the last atom; speculative, no guard branch).
    {
        int natom = atom + 8 < NATOMS ? atom + 8 : NATOMS - 1;
        __builtin_prefetch(p3 + (size_t)natom * (7 * NC), 0, 0);
    }

    // scalars_part0: out[atom][c] = part0[atom][c]; 32 lanes == 32 channels.
    o[lane] = parts[0][lane];

    // Per-l Gram: G = A * A^T with A = P^T (32 x m), P = (m x 32) row-major.
    // 32x32 result as 2x2 grid of 16x16 WMMA tiles; K padded to mult. of 4.
#pragma unroll
    for (int l = 0; l < 4; ++l) {
        const int m = 2 * l + 1;
        const float* __restrict__ p = parts[l];
        float* __restrict__ og = o + NC + l * NC * NC;

#pragma unroll
        for (int i = 0; i < 2; ++i) {          // tile row  (channels c)
#pragma unroll
            for (int j = 0; j < 2; ++j) {      // tile col  (channels d)
                v8f acc = {};
#pragma unroll
                for (int k0 = 0; k0 < m; k0 += 4) {
                    // A-matrix 16x4 f32 layout (ISA 7.12.2):
                    //   v0: lanes 0-15 K=k0+0, lanes 16-31 K=k0+2
                    //   v1: lanes 0-15 K=k0+1, lanes 16-31 K=k0+3
                    // A[c][k] = P[k][c]  (c = 16*i + lm)
                    const int  ka  = k0 + 2 * half;
                    const int  kb  = ka + 1;
                    const bool va  = (ka < m);
                    const bool vb  = (kb < m);
                    // Clamp OOB K to row 0 (always valid) -> unconditional
                    // loads; zero-pad via value select (v_cndmask, no branch).
                    const int  kac = va ? ka : 0;
                    const int  kbc = vb ? kb : 0;

                    const float ax = p[kac * NC + i * 16 + lm];
                    const float ay = p[kbc * NC + i * 16 + lm];
                    // B-matrix 4x16: same K-split across lane halves,
                    // lane lm holds column d = 16*j + lm. B[k][d] = P[k][d].
                    const float bx = p[kac * NC + j * 16 + lm];
                    const float by = p[kbc * NC + j * 16 + lm];

                    v2f A, Bf;
                    A.x  = va ? ax : 0.0f;
                    A.y  = vb ? ay : 0.0f;
                    Bf.x = va ? bx : 0.0f;
                    Bf.y = vb ? by : 0.0f;

                    // D = A x B + C   (full fp32, matches reference einsum)
                    acc = __builtin_amdgcn_wmma_f32_16x16x4_f32(
                        /*neg_a=*/false, A,
                        /*neg_b=*/false, Bf,
                        /*c_mod=*/(short)0, acc,
                        /*reuse_a=*/false, /*reuse_b=*/false);
                }
                // D tile layout: VGPR r, lanes 0-15 -> row 16i+r,   col 16j+lm
                //                       lanes 16-31 -> row 16i+r+8, col 16j+lm
#pragma unroll
                for (int r = 0; r < 8; ++r) {
                    og[(i * 16 + r + 8 * half) * NC + j * 16 + lm] = acc[r];
                }
            }
        }
    }
}

extern "C" void kernel_launch(void* const* d_in, const int* in_sizes, int n_in,
                              void* d_out, int out_size, void* d_ws, size_t ws_size,
                              hipStream_t stream) {
    (void)in_sizes; (void)n_in; (void)out_size; (void)d_ws; (void)ws_size;
    const float* p0 = (const float*)d_in[0];
    const float* p1 = (const float*)d_in[1];
    const float* p2 = (const float*)d_in[2];
    const float* p3 = (const float*)d_in[3];
    float* out = (float*)d_out;

    dim3 grid(NATOMS / 8);   // 2048 blocks, 8 wave32 atoms each
    dim3 block(256);
    so3_gram_wmma_kernel<<<grid, block, 0, stream>>>(p0, p1, p2, p3, out);
}